// QuantumAttentionLayer_87205015978181
// MI455X (gfx1250) — compile-verified
//
#include <hip/hip_runtime.h>
#include <hip/hip_bf16.h>

#define NQ 4
#define DIM 16
#define BB 4
#define SS 512
#define EE 512

typedef __attribute__((ext_vector_type(2))) float v2f;
typedef __attribute__((ext_vector_type(8))) float v8f;

// ---------------------------------------------------------------------------
// K0: (a) build M[16][16] from quantum_weights; (b) pack [Wq|Wk|Wv] -> Wp[512][16]
// (cols 12..15 zero) so k1's WMMA loop is branch-free.
// Threads 0..15 simulate the fixed circuit (2 layers of Rot + CNOT ring) on
// basis state |t>, giving column t of the 16x16 unitary U. Then threads 0..255
// compute M[n][n'] = sum_m zvec[m]*(Ure[m][n]Ure[m][n'] + Uim[m][n]Uim[m][n']),
// zvec[m] = 4 - 2*popcount(m). Wire w acts on bit (3 - w) of the flat index.
// ---------------------------------------------------------------------------
__global__ __launch_bounds__(256) void k0_precompute(const float* __restrict__ qw,
                                                     const float* __restrict__ Wq,
                                                     const float* __restrict__ Wk,
                                                     const float* __restrict__ Wv,
                                                     float* __restrict__ Mout,
                                                     float* __restrict__ Wp) {
  __shared__ float Ure[DIM][DIM];
  __shared__ float Uim[DIM][DIM];
  int t = threadIdx.x;

  // ---- pack Wp[512][16]: cols 0-3 Wq, 4-7 Wk, 8-11 Wv, 12-15 zero ----
  for (int idx = t; idx < EE * 16; idx += 256) {
    int k = idx >> 4, n = idx & 15;
    float v;
    if (n < 4)       v = Wq[k * 4 + n];
    else if (n < 8)  v = Wk[k * 4 + (n - 4)];
    else if (n < 12) v = Wv[k * 4 + (n - 8)];
    else             v = 0.0f;
    Wp[idx] = v;
  }

  if (t < DIM) {
    float sre[DIM], sim[DIM];
#pragma unroll
    for (int n = 0; n < DIM; ++n) { sre[n] = (n == t) ? 1.0f : 0.0f; sim[n] = 0.0f; }
    for (int layer = 0; layer < 2; ++layer) {
      // Rot(a,b,c) = RZ(c) @ RY(b) @ RX(a) on each wire
      for (int w = 0; w < NQ; ++w) {
        const float* w3 = qw + (layer * NQ + w) * 3;
        float a = w3[0] * 0.5f, b = w3[1] * 0.5f, c = w3[2] * 0.5f;
        float ca = cosf(a), sa = sinf(a);
        float cb = cosf(b), sb = sinf(b);
        float cc = cosf(c), sc = sinf(c);
        // RY@RX entries (before RZ phases):
        float m00r = cb * ca, m00i =  sb * sa;
        float m01r = -sb * ca, m01i = -cb * sa;
        float m10r =  sb * ca, m10i = -cb * sa;
        float m11r =  cb * ca, m11i = -sb * sa;
        // row0 *= ez = (cc - i sc); row1 *= conj(ez) = (cc + i sc)
        float G00r = cc * m00r + sc * m00i, G00i = cc * m00i - sc * m00r;
        float G01r = cc * m01r + sc * m01i, G01i = cc * m01i - sc * m01r;
        float G10r = cc * m10r - sc * m10i, G10i = cc * m10i + sc * m10r;
        float G11r = cc * m11r - sc * m11i, G11i = cc * m11i + sc * m11r;
        int mask = 8 >> w;  // wire w -> bit (3-w)
#pragma unroll
        for (int n = 0; n < DIM; ++n) {
          if (n & mask) continue;
          int n1 = n | mask;
          float a0r = sre[n], a0i = sim[n], a1r = sre[n1], a1i = sim[n1];
          sre[n]  = G00r * a0r - G00i * a0i + G01r * a1r - G01i * a1i;
          sim[n]  = G00r * a0i + G00i * a0r + G01r * a1i + G01i * a1r;
          sre[n1] = G10r * a0r - G10i * a0i + G11r * a1r - G11i * a1i;
          sim[n1] = G10r * a0i + G10i * a0r + G11r * a1i + G11i * a1r;
        }
      }
      // CNOT ring (0,1),(1,2),(2,3),(3,0)
      const int ctrls[4] = {0, 1, 2, 3};
      const int tgts[4]  = {1, 2, 3, 0};
#pragma unroll
      for (int e = 0; e < 4; ++e) {
        int cm = 8 >> ctrls[e], tm = 8 >> tgts[e];
        float tr[DIM], ti[DIM];
#pragma unroll
        for (int n = 0; n < DIM; ++n) {
          int src = (n & cm) ? (n ^ tm) : n;
          tr[n] = sre[src]; ti[n] = sim[src];
        }
#pragma unroll
        for (int n = 0; n < DIM; ++n) { sre[n] = tr[n]; sim[n] = ti[n]; }
      }
    }
#pragma unroll
    for (int m = 0; m < DIM; ++m) { Ure[m][t] = sre[m]; Uim[m][t] = sim[m]; }
  }
  __syncthreads();
  {
    int n = t >> 4, np = t & 15;
    float acc = 0.0f;
#pragma unroll
    for (int m = 0; m < DIM; ++m) {
      float z = 4.0f - 2.0f * (float)__popc(m);
      acc += z * (Ure[m][n] * Ure[m][np] + Uim[m][n] * Uim[m][np]);
    }
    Mout[t] = acc;
  }
}

// ---------------------------------------------------------------------------
// K1: qkv projection via chained V_WMMA_F32_16X16X4_F32.
// x[2048,512] @ Wp[512,16]; one wave per 16-row tile; 128 K-steps of K=4
// accumulating into one 16x16 f32 tile. Branch-free inner loop, unrolled x4.
// Output qkv[row][0..3]=tanh(q), [4..7]=tanh(k), [8..11]=v.
// ---------------------------------------------------------------------------
__global__ __launch_bounds__(32) void k1_qkv(const float* __restrict__ x,
                                             const float* __restrict__ Wp,
                                             const float* __restrict__ bq,
                                             const float* __restrict__ bk,
                                             const float* __restrict__ bv,
                                             float* __restrict__ qkv) {
  int lane = threadIdx.x;
  int r0 = blockIdx.x * 16;      // row tile over 2048 rows
  int m  = lane & 15;            // A: M index == B: N index
  int kh = (lane >> 4) * 2;      // K half: 0 or 2
  const float* xrow = x + (size_t)(r0 + m) * EE;
  v8f c = {};
#pragma unroll 4
  for (int k0 = 0; k0 < EE; k0 += 4) {
    int ka = k0 + kh;
    v2f a = *(const v2f*)(xrow + ka);          // global_load_b64
    v2f bm;
    bm.x = Wp[ka * 16 + m];
    bm.y = Wp[ka * 16 + 16 + m];
    c = __builtin_amdgcn_wmma_f32_16x16x4_f32(false, a, false, bm, (short)0, c, false, false);
  }
  int rowAdd = (lane >> 4) * 8;
  int col = lane & 15;
#pragma unroll
  for (int r = 0; r < 8; ++r) {
    int row = r0 + r + rowAdd;
    if (col < 12) {
      float v = c[r];
      if (col < 4)      { v += bq[col];     v = tanhf(v); }
      else if (col < 8) { v += bk[col - 4]; v = tanhf(v); }
      else              { v += bv[col - 8]; }
      qkv[row * 12 + col] = v;
    }
  }
}

// ---------------------------------------------------------------------------
// K2: per (b,i) row: score(j) = 0.5 * amp^T M amp with
// amp = f0 x f1 x f2 x f3, f_w = (cos(th/2), sin(th/2)), th = tanh(q_w + k_jw).
// Block softmax over j (wave32 shuffle reductions + one cross-wave LDS pass),
// write attn row, and reduce attended = sum_j p*v_j.
// ---------------------------------------------------------------------------
#define NWAVES (SS / 32)

__device__ __forceinline__ float waveMax(float v) {
#pragma unroll
  for (int o = 16; o > 0; o >>= 1) v = fmaxf(v, __shfl_xor(v, o, 32));
  return v;
}
__device__ __forceinline__ float waveSum(float v) {
#pragma unroll
  for (int o = 16; o > 0; o >>= 1) v += __shfl_xor(v, o, 32);
  return v;
}

__global__ __launch_bounds__(512) void k2_scores(const float* __restrict__ qkv,
                                                 const float* __restrict__ M,
                                                 float* __restrict__ attn,
                                                 float* __restrict__ attended) {
  __shared__ float Ms[256];
  __shared__ float wred[NWAVES];
  __shared__ float4 wred4[NWAVES];
  int row = blockIdx.x;          // b*S + i
  int b = row / SS;
  int j = threadIdx.x;           // 0..511
  int laneid = j & 31, waveid = j >> 5;
  if (j < 256) Ms[j] = M[j];
  __syncthreads();

  const float* qrow = qkv + row * 12;
  const float* krow = qkv + (b * SS + j) * 12 + 4;
  float f[NQ][2];
#pragma unroll
  for (int w = 0; w < NQ; ++w) {
    float th = tanhf(qrow[w] + krow[w]);
    float sn, cs;
    sincosf(0.5f * th, &sn, &cs);
    f[w][0] = cs; f[w][1] = sn;
  }
  float amp[DIM];
#pragma unroll
  for (int n = 0; n < DIM; ++n)
    amp[n] = f[0][(n >> 3) & 1] * f[1][(n >> 2) & 1] * f[2][(n >> 1) & 1] * f[3][n & 1];
  float score = 0.0f;
#pragma unroll
  for (int n = 0; n < DIM; ++n) {
    float y = 0.0f;
#pragma unroll
    for (int np = 0; np < DIM; ++np) y += Ms[n * 16 + np] * amp[np];
    score += amp[n] * y;
  }
  score *= 0.5f;  // scores / sqrt(N_QUBITS)

  // ---- block max ----
  float wm = waveMax(score);
  if (laneid == 0) wred[waveid] = wm;
  __syncthreads();
  if (j == 0) {
    float mm = wred[0];
#pragma unroll
    for (int i = 1; i < NWAVES; ++i) mm = fmaxf(mm, wred[i]);
    wred[0] = mm;
  }
  __syncthreads();
  float mx = wred[0];

  // ---- block sum of exp ----
  float e = expf(score - mx);
  float ws = waveSum(e);
  __syncthreads();            // wred[0] consumed by all before overwrite
  if (laneid == 0) wred[waveid] = ws;
  __syncthreads();
  if (j == 0) {
    float ss = wred[0];
#pragma unroll
    for (int i = 1; i < NWAVES; ++i) ss += wred[i];
    wred[0] = ss;
  }
  __syncthreads();
  float p = e / wred[0];
  attn[row * SS + j] = p;

  // ---- attended[row] = sum_j p * v_j ----
  const float* vrow = qkv + (b * SS + j) * 12 + 8;
  float a0 = waveSum(p * vrow[0]);
  float a1 = waveSum(p * vrow[1]);
  float a2 = waveSum(p * vrow[2]);
  float a3 = waveSum(p * vrow[3]);
  if (laneid == 0) wred4[waveid] = make_float4(a0, a1, a2, a3);
  __syncthreads();
  if (j == 0) {
    float4 s = wred4[0];
#pragma unroll
    for (int i = 1; i < NWAVES; ++i) {
      s.x += wred4[i].x; s.y += wred4[i].y; s.z += wred4[i].z; s.w += wred4[i].w;
    }
    attended[row * 4 + 0] = s.x;
    attended[row * 4 + 1] = s.y;
    attended[row * 4 + 2] = s.z;
    attended[row * 4 + 3] = s.w;
  }
}

// ---------------------------------------------------------------------------
// K3: out = attended[2048,4] @ Wout[4,512] + bout.
// K=4 matches one V_WMMA_F32_16X16X4_F32 per 16x16 tile. One wave per tile.
// ---------------------------------------------------------------------------
__global__ __launch_bounds__(32) void k3_out(const float* __restrict__ attended,
                                             const float* __restrict__ Wout,
                                             const float* __restrict__ bout,
                                             float* __restrict__ out) {
  int lane = threadIdx.x;
  int c0 = blockIdx.x * 16;     // 0..31 col tiles
  int r0 = blockIdx.y * 16;     // 0..127 row tiles
  int m  = lane & 15;
  int kh = (lane >> 4) * 2;
  v2f a = *(const v2f*)(attended + (r0 + m) * 4 + kh);   // global_load_b64
  v2f bm;
  bm.x = Wout[kh * EE + c0 + m];
  bm.y = Wout[(kh + 1) * EE + c0 + m];
  v8f c = {};
  c = __builtin_amdgcn_wmma_f32_16x16x4_f32(false, a, false, bm, (short)0, c, false, false);
  int rowAdd = (lane >> 4) * 8;
  int col = c0 + (lane & 15);
#pragma unroll
  for (int r = 0; r < 8; ++r) {
    int row = r0 + r + rowAdd;
    out[row * EE + col] = c[r] + bout[col];
  }
}

// ---------------------------------------------------------------------------
extern "C" void kernel_launch(void* const* d_in, const int* in_sizes, int n_in,
                              void* d_out, int out_size, void* d_ws, size_t ws_size,
                              hipStream_t stream) {
  const float* x    = (const float*)d_in[0];
  const float* Wq   = (const float*)d_in[1];
  const float* bq   = (const float*)d_in[2];
  const float* Wk   = (const float*)d_in[3];
  const float* bk   = (const float*)d_in[4];
  const float* Wv   = (const float*)d_in[5];
  const float* bv   = (const float*)d_in[6];
  const float* qw   = (const float*)d_in[7];
  const float* Wout = (const float*)d_in[8];
  const float* bout = (const float*)d_in[9];

  float* out  = (float*)d_out;                       // [B,S,E]
  float* attn = out + (size_t)BB * SS * EE;          // [B,S,S]

  float* ws       = (float*)d_ws;
  float* Mws      = ws;                              // 256 floats
  float* Wp       = ws + 256;                        // 512*16 floats
  float* qkv      = Wp + (size_t)EE * 16;            // 2048*12 floats
  float* attended = qkv + (size_t)BB * SS * 12;      // 2048*4 floats

  k0_precompute<<<1, 256, 0, stream>>>(qw, Wq, Wk, Wv, Mws, Wp);
  k1_qkv<<<(BB * SS) / 16, 32, 0, stream>>>(x, Wp, bq, bk, bv, qkv);
  k2_scores<<<BB * SS, SS, 0, stream>>>(qkv, Mws, attn, attended);
  k3_out<<<dim3(EE / 16, (BB * SS) / 16), 32, 0, stream>>>(attended, Wout, bout, out);
}